// OfficialPostProcessHOI_15831249453186
// MI455X (gfx1250) — compile-verified
//
#include <hip/hip_runtime.h>
#include <hip/hip_bf16.h>
#include <math.h>

#define BB 512
#define QQ 100
#define CC 81
#define VV 117
#define THRES 0.7f

#define ROWS (BB * QQ)            // 51200
#define WAVES_PER_BLOCK 8         // 256 threads
#define ROWS_PER_WAVE 8
#define NBLK (ROWS / (WAVES_PER_BLOCK * ROWS_PER_WAVE))  // 800

// flat output offsets (tuple concatenated in return order, all float32)
#define HOI_OFF  ((size_t)0)
#define LAB_OFF  ((size_t)BB * QQ * VV)
#define SUB_OFF  (LAB_OFF + (size_t)BB * QQ)
#define OBJ_OFF  (SUB_OFF + (size_t)BB * QQ * 4)
#define KEEP_OFF (OBJ_OFF + (size_t)BB * QQ * 4)

#if defined(__AMDGCN__) && __has_builtin(__builtin_amdgcn_global_load_async_to_lds_b32) && __has_builtin(__builtin_amdgcn_s_wait_asynccnt)
#define USE_ASYNC_LDS 1
#else
#define USE_ASYNC_LDS 0
#endif

// ---------------------------------------------------------------------------
// Kernel 1: 256 threads (8 wave32), each wave processes 8 (b,q) rows.
//  - softmax over 81 obj logits (max/sum via butterfly shuffles)
//  - obj_score / obj_label = max/argmax over first 80 probabilities
//  - hoi_scores[v] = sigmoid(verb[v]) * obj_score * correct_mat[v, label]
//  - per-row max -> workspace (NMS scores); label -> workspace + output
//  - cxcywh -> xyxy box scaling (lanes 0/1)
// correct_mat (117x81 = 37.9 KB) staged once per block to LDS via the CDNA5
// async global->LDS path (amortized over 64 rows/block: ~30 MB total traffic).
// ---------------------------------------------------------------------------
__global__ __launch_bounds__(256) void hoi_row_kernel(
    const float* __restrict__ obj_logits,   // (B,Q,81)
    const float* __restrict__ verb_logits,  // (B,Q,117)
    const float* __restrict__ sub_boxes,    // (B,Q,4)
    const float* __restrict__ obj_boxes,    // (B,Q,4)
    const float* __restrict__ correct_mat,  // (117,81)
    const int* __restrict__ target_sizes,   // (B,2) int32 (h,w)
    float* __restrict__ out,
    float* __restrict__ ws_scores,          // (B*Q)
    int* __restrict__ ws_labels)            // (B*Q)
{
  __shared__ float cm[VV * CC];
  const int tid = threadIdx.x;

#if USE_ASYNC_LDS
  for (int i = tid; i < VV * CC; i += 256) {
    __builtin_amdgcn_global_load_async_to_lds_b32(
        (__attribute__((address_space(1))) int*)(const_cast<float*>(correct_mat + i)),
        (__attribute__((address_space(3))) int*)(&cm[i]), 0, 0);
  }
  __builtin_amdgcn_s_wait_asynccnt(0);
#else
  for (int i = tid; i < VV * CC; i += 256) cm[i] = correct_mat[i];
#endif
  __syncthreads();

  const int lane = tid & 31;
  const int gwave = blockIdx.x * WAVES_PER_BLOCK + (tid >> 5);

  for (int r = 0; r < ROWS_PER_WAVE; ++r) {
    const int row = gwave * ROWS_PER_WAVE + r;   // grid sized exactly: B*Q rows
    const int b = row / QQ;

    const float* lg = obj_logits + (size_t)row * CC;
    const float* vb = verb_logits + (size_t)row * VV;
    __builtin_prefetch(vb + lane, 0, 0);  // global_prefetch_b8 on the verb row

    // 81 logits: lanes hold idx {lane, lane+32, lane+64}
    const float x0 = lg[lane];
    const float x1 = lg[lane + 32];
    const float x2 = (lane < CC - 64) ? lg[lane + 64] : -__builtin_inff();

    // max over all 81 (softmax stability)
    float mall = fmaxf(fmaxf(x0, x1), x2);
    for (int off = 16; off; off >>= 1) mall = fmaxf(mall, __shfl_xor(mall, off, 32));

    // softmax denominator over all 81
    float s = expf(x0 - mall) + expf(x1 - mall);
    if (lane < CC - 64) s += expf(x2 - mall);
    for (int off = 16; off; off >>= 1) s += __shfl_xor(s, off, 32);

    // max/argmax over first 80 (exclude idx 80 == lane 16 of x2); ties -> lower idx
    float mv = x0;
    int mi = lane;
    if (x1 > mv) { mv = x1; mi = lane + 32; }
    if (lane < 16 && x2 > mv) { mv = x2; mi = lane + 64; }
    for (int off = 16; off; off >>= 1) {
      const float ov = __shfl_xor(mv, off, 32);
      const int oi = __shfl_xor(mi, off, 32);
      if (ov > mv || (ov == mv && oi < mi)) { mv = ov; mi = oi; }
    }
    const float obj_score = expf(mv - mall) / s;
    const int label = mi;

    // hoi scores + row max
    float* ho = out + HOI_OFF + (size_t)row * VV;
    float hmax = -__builtin_inff();
    for (int v = lane; v < VV; v += 32) {
      const float sig = 1.0f / (1.0f + expf(-vb[v]));
      const float h = sig * obj_score * cm[v * CC + label];
      ho[v] = h;
      hmax = fmaxf(hmax, h);
    }
    for (int off = 16; off; off >>= 1) hmax = fmaxf(hmax, __shfl_xor(hmax, off, 32));

    if (lane == 0) {
      ws_scores[row] = hmax;
      ws_labels[row] = label;
      out[LAB_OFF + row] = (float)label;
    }

    // box conversion + scale: lane 0 = sub box, lane 1 = obj box
    if (lane < 2) {
      const float* bp = ((lane == 0) ? sub_boxes : obj_boxes) + (size_t)row * 4;
      const float cx = bp[0], cy = bp[1], w = bp[2], h = bp[3];
      const float ih = (float)target_sizes[b * 2 + 0];
      const float iw = (float)target_sizes[b * 2 + 1];
      float* op = out + ((lane == 0) ? SUB_OFF : OBJ_OFF) + (size_t)row * 4;
      op[0] = (cx - 0.5f * w) * iw;
      op[1] = (cy - 0.5f * h) * ih;
      op[2] = (cx + 0.5f * w) * iw;
      op[3] = (cy + 0.5f * h) * ih;
    }
  }
}

// ---------------------------------------------------------------------------
// Kernel 2: one block (128 threads) per image. Stable descending rank-sort of
// 100 scores, then sequential greedy suppression (parallel over j).
// ovr(i,j) = iou_sub^ALPHA * iou_obj^BETA = iou_sub * sqrt(iou_obj).
// ---------------------------------------------------------------------------
__global__ __launch_bounds__(128) void hoi_nms_kernel(
    float* __restrict__ out,
    const float* __restrict__ ws_scores,
    const int* __restrict__ ws_labels)
{
  __shared__ float ssc[QQ];
  __shared__ int slab[QQ];
  __shared__ int sorder[QQ];
  __shared__ int slabs[QQ];     // sorted labels
  __shared__ float sb[QQ * 4];  // sorted sub boxes
  __shared__ float ob[QQ * 4];  // sorted obj boxes
  __shared__ float asub[QQ], aobj[QQ];
  __shared__ int supp[QQ];

  const int b = blockIdx.x;
  const int tid = threadIdx.x;

  for (int i = tid; i < QQ; i += 128) {
    ssc[i] = ws_scores[b * QQ + i];
    slab[i] = ws_labels[b * QQ + i];
  }
  __syncthreads();

  // stable rank: equivalent to argsort(-scores, stable)
  for (int i = tid; i < QQ; i += 128) {
    const float si = ssc[i];
    int r = 0;
    for (int j = 0; j < QQ; ++j) {
      const float sj = ssc[j];
      r += (sj > si) || ((sj == si) && (j < i));
    }
    sorder[r] = i;
  }
  __syncthreads();

  // gather sorted boxes/labels, precompute areas (+1 convention)
  for (int k = tid; k < QQ; k += 128) {
    const int i = sorder[k];
    const float* sp = out + SUB_OFF + (size_t)(b * QQ + i) * 4;
    const float* op = out + OBJ_OFF + (size_t)(b * QQ + i) * 4;
    sb[k * 4 + 0] = sp[0]; sb[k * 4 + 1] = sp[1];
    sb[k * 4 + 2] = sp[2]; sb[k * 4 + 3] = sp[3];
    ob[k * 4 + 0] = op[0]; ob[k * 4 + 1] = op[1];
    ob[k * 4 + 2] = op[2]; ob[k * 4 + 3] = op[3];
    asub[k] = (sp[2] - sp[0] + 1.0f) * (sp[3] - sp[1] + 1.0f);
    aobj[k] = (op[2] - op[0] + 1.0f) * (op[3] - op[1] + 1.0f);
    slabs[k] = slab[i];
    supp[k] = 0;
  }
  __syncthreads();

  for (int i = 0; i < QQ - 1; ++i) {
    if (!supp[i]) {  // uniform branch (all threads read same LDS word)
      const float sx1 = sb[i * 4 + 0], sy1 = sb[i * 4 + 1];
      const float sx2 = sb[i * 4 + 2], sy2 = sb[i * 4 + 3];
      const float ox1 = ob[i * 4 + 0], oy1 = ob[i * 4 + 1];
      const float ox2 = ob[i * 4 + 2], oy2 = ob[i * 4 + 3];
      const float sai = asub[i], oai = aobj[i];
      const int li = slabs[i];
      for (int j = i + 1 + tid; j < QQ; j += 128) {
        if (slabs[j] == li) {
          // sub-box IoU
          float xx1 = fmaxf(sx1, sb[j * 4 + 0]);
          float yy1 = fmaxf(sy1, sb[j * 4 + 1]);
          float xx2 = fminf(sx2, sb[j * 4 + 2]);
          float yy2 = fminf(sy2, sb[j * 4 + 3]);
          float w = fmaxf(0.0f, xx2 - xx1 + 1.0f);
          float h = fmaxf(0.0f, yy2 - yy1 + 1.0f);
          float inter = w * h;
          const float iou_s = inter / (sai + asub[j] - inter);
          // obj-box IoU
          xx1 = fmaxf(ox1, ob[j * 4 + 0]);
          yy1 = fmaxf(oy1, ob[j * 4 + 1]);
          xx2 = fminf(ox2, ob[j * 4 + 2]);
          yy2 = fminf(oy2, ob[j * 4 + 3]);
          w = fmaxf(0.0f, xx2 - xx1 + 1.0f);
          h = fmaxf(0.0f, yy2 - yy1 + 1.0f);
          inter = w * h;
          const float iou_o = inter / (oai + aobj[j] - inter);
          if (iou_s * sqrtf(iou_o) > THRES) supp[j] = 1;
        }
      }
    }
    __syncthreads();
  }

  for (int k = tid; k < QQ; k += 128) {
    out[KEEP_OFF + b * QQ + sorder[k]] = supp[k] ? 0.0f : 1.0f;
  }
}

extern "C" void kernel_launch(void* const* d_in, const int* in_sizes, int n_in,
                              void* d_out, int out_size, void* d_ws, size_t ws_size,
                              hipStream_t stream) {
  const float* obj_logits = (const float*)d_in[0];
  const float* verb_logits = (const float*)d_in[1];
  const float* sub_boxes = (const float*)d_in[2];
  const float* obj_boxes = (const float*)d_in[3];
  const float* correct_mat = (const float*)d_in[4];
  const int* target_sizes = (const int*)d_in[5];
  float* out = (float*)d_out;
  float* ws_scores = (float*)d_ws;
  int* ws_labels = (int*)((char*)d_ws + (size_t)BB * QQ * sizeof(float));

  // Kernel 1: 51,200 rows; 8 waves/block x 8 rows/wave -> 800 blocks.
  hoi_row_kernel<<<NBLK, 256, 0, stream>>>(
      obj_logits, verb_logits, sub_boxes, obj_boxes, correct_mat, target_sizes,
      out, ws_scores, ws_labels);

  // Kernel 2: one block per image.
  hoi_nms_kernel<<<BB, 128, 0, stream>>>(out, ws_scores, ws_labels);
}